// HSIC_weight_52690658788023
// MI455X (gfx1250) — compile-verified
//
#include <hip/hip_runtime.h>
#include <hip/hip_bf16.h>

// HSIC-style pairwise loss, algebraically reduced:
//   loss = sum_{i<j} || G_ij - (1/n) s_i s_j^T ||_F^2 / (n-1)^2
//   G_ij[p,q] = sum_a w_a^4 F[a, i*e+p] F[a, j*e+q]   (w^4 folded into B fragments)
//   s[c]      = sum_a w_a^2 F[a, c]
// Heavy work: 45 GEMMs of 256x256xK4096 in fp32 -> V_WMMA_F32_16X16X4_F32,
// panels + w^4 slices staged via GLOBAL_LOAD_ASYNC_TO_LDS_B128, double buffered.

#define NN      4096
#define EMBD    256
#define NCHUNK  10
#define COLS    (NCHUNK * EMBD)   // 2560
#define NPAIR   45                // i<j pairs

#define BM 128
#define BN 128
#define BK 32
#define LS 136                    // padded LDS row stride (dwords), multiple of 4 for b128
#define NPANELS (NN / BK)         // 128
#define THREADS 128               // 4 wave32s, each owns a 64x64 tile
#define IPT 16                    // min async b128 instructions per wave per panel (A+B)

typedef __attribute__((ext_vector_type(2))) float v2f;
typedef __attribute__((ext_vector_type(8))) float v8f;
typedef __attribute__((ext_vector_type(4))) int   v4i;

// Address-space qualified element types for the async-copy builtin
// (param 0 is v4i addrspace(1)*, param 1 is v4i addrspace(3)*).
typedef __attribute__((address_space(1))) v4i glob_v4i;
typedef __attribute__((address_space(3))) v4i lds_v4i;

// Workspace layout (floats): s colsums, w^2, w^4
#define WS_S  0
#define WS_W2 COLS
#define WS_W4 (COLS + NN)

// ---------------------------------------------------------------------------
// CDNA5 async global->LDS copy (ASYNCcnt path), with sync fallback.
// ---------------------------------------------------------------------------
#if defined(__HIP_DEVICE_COMPILE__) && defined(__gfx1250__) && \
    __has_builtin(__builtin_amdgcn_global_load_async_to_lds_b128)
#define HAVE_ASYNC 1
#else
#define HAVE_ASYNC 0
#endif

__device__ __forceinline__ void copy16(const float* g, float* l) {
#if HAVE_ASYNC
    __builtin_amdgcn_global_load_async_to_lds_b128(
        (glob_v4i*)g, (lds_v4i*)l, 0, 0);
#else
    const float4 v = *(const float4*)g;
    *(float4*)l = v;
#endif
}

template <int N>
__device__ __forceinline__ void wait_async() {
#if HAVE_ASYNC
#if __has_builtin(__builtin_amdgcn_s_wait_asynccnt)
    __builtin_amdgcn_s_wait_asynccnt(N);
#else
    asm volatile("s_wait_asynccnt %0" ::"i"(N));
#endif
#endif
}

// ---------------------------------------------------------------------------
// Prep: zero colsum accumulators + loss output, precompute w^2 and w^4.
// ---------------------------------------------------------------------------
__global__ void hsic_prep_kernel(const float* __restrict__ w,
                                 float* __restrict__ ws,
                                 float* __restrict__ out) {
    int t = blockIdx.x * blockDim.x + threadIdx.x;
    if (t < COLS) ws[WS_S + t] = 0.0f;
    if (t < NN) {
        float v  = w[t];
        float v2 = v * v;
        ws[WS_W2 + t] = v2;
        ws[WS_W4 + t] = v2 * v2;
    }
    if (t == 0) out[0] = 0.0f;
}

// ---------------------------------------------------------------------------
// Column sums of D = diag(w^2) * F  ->  s[0..COLS)
// ---------------------------------------------------------------------------
__global__ void hsic_colsum_kernel(const float* __restrict__ F,
                                   float* __restrict__ ws) {
    const float* w2 = ws + WS_W2;
    int col = blockIdx.x * 256 + threadIdx.x;
    int r0  = blockIdx.y * 256;
    float acc = 0.0f;
    #pragma unroll 4
    for (int r = 0; r < 256; ++r) {
        int a = r0 + r;
        acc = __builtin_fmaf(w2[a], F[(size_t)a * COLS + col], acc);
    }
    atomicAdd(&ws[WS_S + col], acc);
}

// ---------------------------------------------------------------------------
// Stage one BKx128 panel of F columns [col0, col0+128) for A and B plus the
// BK-float w^4 slice into LDS. 1024 b128 slots per matrix / 128 threads = 8
// per thread per matrix; w^4 slice = 8 b128 slots issued by threads 0..7
// (all in wave 0, so IPT=16 remains a valid lower bound for every wave).
// ---------------------------------------------------------------------------
__device__ __forceinline__ void stage_panel(const float* __restrict__ F,
                                            const float* __restrict__ w4,
                                            int a0, int colA0, int colB0,
                                            float* bufA, float* bufB, float* bufW,
                                            int tid) {
    #pragma unroll
    for (int rep = 0; rep < 8; ++rep) {
        int linear = tid + rep * THREADS;     // 0..1023
        int row = linear >> 5;                // 0..31
        int c4  = (linear & 31) << 2;         // dword column 0..124
        const float* g = F + (size_t)(a0 + row) * COLS;
        copy16(g + colA0 + c4, bufA + row * LS + c4);
        copy16(g + colB0 + c4, bufB + row * LS + c4);
    }
    if (tid < BK / 4)
        copy16(w4 + a0 + tid * 4, bufW + tid * 4);
}

// ---------------------------------------------------------------------------
// Main kernel: per (pair, quadrant) compute a 128x128 tile of G_ij over
// K = 4096 with fp32 WMMA. 4 wave32s, wave tile 64x64 = 4x4 accumulators.
// Double-buffered async LDS staging. Epilogue folds the rank-1 mean term,
// squares, reduces, atomically accumulates the scalar loss.
// ---------------------------------------------------------------------------
__launch_bounds__(THREADS)
__global__ void hsic_gram_kernel(const float* __restrict__ F,
                                 const float* __restrict__ ws,
                                 float* __restrict__ out) {
    __shared__ float ldsA[2][BK * LS];
    __shared__ float ldsB[2][BK * LS];
    __shared__ float ldsW[2][BK];
    __shared__ float wred[4];

    const float* s  = ws + WS_S;
    const float* w4 = ws + WS_W4;

    // Decode chunk pair (ci < cj) from blockIdx.y in [0, 45)
    int ci = 0, cj = 1;
    {
        int t = blockIdx.y;
        #pragma unroll
        for (int i = 0; i < NCHUNK; ++i) {
            int cnt = NCHUNK - 1 - i;
            if (t < cnt) { ci = i; cj = i + 1 + t; break; }
            t -= cnt;
        }
    }
    const int qm = blockIdx.x & 1;               // quadrant of the 256x256 pair output
    const int qn = blockIdx.x >> 1;
    const int colA0 = ci * EMBD + qm * BM;       // F columns feeding the A panel
    const int colB0 = cj * EMBD + qn * BN;       // F columns feeding the B panel

    const int tid  = threadIdx.x;
    const int lane = tid & 31;
    const int wave = tid >> 5;
    const int wm = (wave & 1) * 64;              // wave M offset inside 128
    const int wn = (wave >> 1) * 64;             // wave N offset inside 128
    // f32 WMMA fragment addressing (ISA 7.12.2): lanes 0-15 hold K=0,1;
    // lanes 16-31 hold K=2,3; M/N index = lane % 16.
    const int kk = (lane >> 4) << 1;             // 0 or 2
    const int ml = lane & 15;

    v8f acc[4][4];
    #pragma unroll
    for (int mt = 0; mt < 4; ++mt)
        #pragma unroll
        for (int nt = 0; nt < 4; ++nt)
            acc[mt][nt] = (v8f){};

    int cur = 0;
    stage_panel(F, w4, 0, colA0, colB0, ldsA[0], ldsB[0], ldsW[0], tid);

    for (int p = 0; p < NPANELS; ++p) {
        const int a0 = p * BK;
        if (p + 1 < NPANELS) {
            stage_panel(F, w4, a0 + BK, colA0, colB0,
                        ldsA[cur ^ 1], ldsB[cur ^ 1], ldsW[cur ^ 1], tid);
            wait_async<IPT>();   // FIFO: panel p fully resident in LDS
        } else {
            wait_async<0>();
        }
        __syncthreads();

        const float* bufA = ldsA[cur];
        const float* bufB = ldsB[cur];
        const float* bufW = ldsW[cur];

        #pragma unroll
        for (int k = 0; k < BK; k += 4) {
            // per-K w^4 factors for this lane's two K slices (broadcast ds_load_b64)
            const v2f w4p = *(const v2f*)&bufW[k + kk];
            v2f bfrag[4];
            #pragma unroll
            for (int nt = 0; nt < 4; ++nt) {
                int c = wn + nt * 16 + ml;
                bfrag[nt].x = bufB[(k + kk) * LS + c] * w4p.x;
                bfrag[nt].y = bufB[(k + kk + 1) * LS + c] * w4p.y;
            }
            #pragma unroll
            for (int mt = 0; mt < 4; ++mt) {
                int c = wm + mt * 16 + ml;
                v2f afrag;
                afrag.x = bufA[(k + kk) * LS + c];
                afrag.y = bufA[(k + kk + 1) * LS + c];
                #pragma unroll
                for (int nt = 0; nt < 4; ++nt) {
                    acc[mt][nt] = __builtin_amdgcn_wmma_f32_16x16x4_f32(
                        false, afrag, false, bfrag[nt],
                        (short)0, acc[mt][nt], false, false);
                }
            }
        }
        __syncthreads();
        cur ^= 1;
    }

    // ---- epilogue: (G - s_i s_j^T / n)^2, reduce, accumulate scalar loss ----
    const float inv_n = 1.0f / (float)NN;
    const int mhalf = (lane >> 4) << 3;          // C/D layout: lanes 16-31 hold M+8
    float sumsq = 0.0f;
    #pragma unroll
    for (int mt = 0; mt < 4; ++mt) {
        #pragma unroll
        for (int nt = 0; nt < 4; ++nt) {
            int qcol = colB0 + wn + nt * 16 + ml;                 // s_j index
            float sq = s[qcol] * inv_n;
            #pragma unroll
            for (int r = 0; r < 8; ++r) {
                int pcol = colA0 + wm + mt * 16 + mhalf + r;      // s_i index
                float v = acc[mt][nt][r] - s[pcol] * sq;
                sumsq = __builtin_fmaf(v, v, sumsq);
            }
        }
    }

    // wave32 reduction, then cross-wave via LDS
    #pragma unroll
    for (int off = 16; off > 0; off >>= 1)
        sumsq += __shfl_down(sumsq, off, 32);
    if (lane == 0) wred[wave] = sumsq;
    __syncthreads();
    if (tid == 0) {
        float tot = wred[0] + wred[1] + wred[2] + wred[3];
        const float scale = 1.0f / ((float)(NN - 1) * (float)(NN - 1));
        atomicAdd(out, tot * scale);
    }
}

// ---------------------------------------------------------------------------
extern "C" void kernel_launch(void* const* d_in, const int* in_sizes, int n_in,
                              void* d_out, int out_size, void* d_ws, size_t ws_size,
                              hipStream_t stream) {
    const float* F = (const float*)d_in[0];   // final_readout [4096, 2560]
    const float* w = (const float*)d_in[1];   // weight        [4096, 1]
    float* out = (float*)d_out;               // scalar loss
    float* ws  = (float*)d_ws;                // >= (COLS + 2*NN) * 4 bytes

    hsic_prep_kernel<<<dim3((NN + 255) / 256), dim3(256), 0, stream>>>(w, ws, out);
    hsic_colsum_kernel<<<dim3(COLS / 256, NN / 256), dim3(256), 0, stream>>>(F, ws);
    hsic_gram_kernel<<<dim3(4, NPAIR), dim3(THREADS), 0, stream>>>(F, ws, out);
}